// MSSSIMLoss_14456859918789
// MI455X (gfx1250) — compile-verified
//
#include <hip/hip_runtime.h>
#include <math.h>

typedef float v2f __attribute__((ext_vector_type(2)));
typedef float v8f __attribute__((ext_vector_type(8)));

#define BC_TOTAL 48          // 16 batch * 3 channels
#define KWIN 11
#define TILE 16
#define PATCH 26             // TILE + KWIN - 1
#define PATCHP 28            // padded K extent (multiple of 4)
#define WAVES_PER_BLOCK 4

// ---------------------------------------------------------------------------
// Per-level SSIM kernel: one wave32 computes one 16x16 tile of the valid
// 11x11 depthwise-Gaussian conv outputs for 5 planes (x, y, x^2, y^2, x*y)
// with V_WMMA_F32_16X16X4_F32 for both separable passes.
// Banded Gaussian weight fragments are hoisted into 7 v2f registers
// (identical for the H-pass B operand and V-pass A operand), so the hot
// loops contain only unconditional ds_loads + plane products + WMMAs.
// ---------------------------------------------------------------------------
__global__ __launch_bounds__(128)
void msssim_level_kernel(const float* __restrict__ X, const float* __restrict__ Y,
                         float* __restrict__ partials, int H, int W) {
    __shared__ float xs[WAVES_PER_BLOCK][PATCH][PATCHP];
    __shared__ float ys[WAVES_PER_BLOCK][PATCH][PATCHP];
    __shared__ float ht[WAVES_PER_BLOCK][5][PATCHP][TILE + 1];
    __shared__ float gwpad[48];   // gwpad[16+d] = g[d] for d in [0,10], else 0

    const int tid  = threadIdx.x;
    const int w    = tid >> 5;        // wave id (wave32)
    const int lane = tid & 31;
    const int half = lane >> 4;       // 0: lanes 0-15, 1: lanes 16-31
    const int l16  = lane & 15;

    if (tid == 0) {
        #pragma unroll
        for (int i = 0; i < 48; ++i) gwpad[i] = 0.f;
        gwpad[16] = 0.00102838f; gwpad[17] = 0.00759875f; gwpad[18] = 0.03600077f;
        gwpad[19] = 0.10936070f; gwpad[20] = 0.21300553f; gwpad[21] = 0.26601172f;
        gwpad[22] = 0.21300553f; gwpad[23] = 0.10936070f; gwpad[24] = 0.03600077f;
        gwpad[25] = 0.00759875f; gwpad[26] = 0.00102838f;
    }

    const int Hc = H - (KWIN - 1);
    const int Wc = W - (KWIN - 1);
    const int T  = (Hc + TILE - 1) / TILE;       // tiles per dim (square imgs)
    const int tilesPerImg = T * T;
    const int total = BC_TOTAL * tilesPerImg;

    int tileFlat = blockIdx.x * WAVES_PER_BLOCK + w;
    const bool active = (tileFlat < total);
    if (!active) tileFlat = 0;                   // redundant compute, no store

    const int bc  = tileFlat / tilesPerImg;
    const int rem = tileFlat % tilesPerImg;
    const int tr  = rem / T;
    const int tc  = rem % T;
    const int baseR = tr * TILE;
    const int baseC = tc * TILE;

    const float* Xb = X + (size_t)bc * H * W;
    const float* Yb = Y + (size_t)bc * H * W;

    // Zero the two K-padding rows of every ht plane (read via zero weights,
    // but must be finite).
    for (int i = lane; i < 5 * 2 * (TILE + 1); i += 32) {
        int p  = i / (2 * (TILE + 1));
        int r  = (i / (TILE + 1)) & 1;
        int c  = i % (TILE + 1);
        ht[w][p][PATCH + r][c] = 0.f;
    }

    // Stage the 26x28 input patches (edge-clamped; K-pad cols get clamped
    // finite values that are multiplied by zero weights).
    for (int i = lane; i < PATCH * PATCHP; i += 32) {
        int r = i / PATCHP, c = i % PATCHP;
        int gr = baseR + r; if (gr > H - 1) gr = H - 1;
        int gc = baseC + c; if (gc > W - 1) gc = W - 1;
        size_t off = (size_t)gr * W + gc;
        xs[w][r][c] = Xb[off];
        ys[w][r][c] = Yb[off];
    }
    __syncthreads();

    // Hoisted banded Gaussian fragments: wfrag[t] serves as B in the H-pass
    // and A in the V-pass (same value g[kg - l16], zero outside the band).
    v2f wfrag[7];
    #pragma unroll
    for (int t = 0; t < 7; ++t) {
        int k0 = 4 * t + 2 * half;
        wfrag[t][0] = gwpad[16 + k0 - l16];
        wfrag[t][1] = gwpad[16 + k0 + 1 - l16];
    }

    // ---- horizontal pass: ht_p(26x16) = In_p(26x28) * Wh(28x16), 5 planes ----
    #pragma unroll
    for (int mc = 0; mc < 2; ++mc) {
        v8f acc0 = {}, acc1 = {}, acc2 = {}, acc3 = {}, acc4 = {};
        const int m    = mc * TILE + l16;
        const int rowi = (m < PATCH) ? m : (PATCH - 1);
        #pragma unroll
        for (int t = 0; t < 7; ++t) {
            const int k0 = 4 * t + 2 * half;
            float px0 = xs[w][rowi][k0];
            float px1 = xs[w][rowi][k0 + 1];
            float py0 = ys[w][rowi][k0];
            float py1 = ys[w][rowi][k0 + 1];
            v2f ax  = {px0, px1};
            v2f ay  = {py0, py1};
            v2f axx = {px0 * px0, px1 * px1};
            v2f ayy = {py0 * py0, py1 * py1};
            v2f axy = {px0 * py0, px1 * py1};
            acc0 = __builtin_amdgcn_wmma_f32_16x16x4_f32(false, ax,  false, wfrag[t], (short)0, acc0, false, false);
            acc1 = __builtin_amdgcn_wmma_f32_16x16x4_f32(false, ay,  false, wfrag[t], (short)0, acc1, false, false);
            acc2 = __builtin_amdgcn_wmma_f32_16x16x4_f32(false, axx, false, wfrag[t], (short)0, acc2, false, false);
            acc3 = __builtin_amdgcn_wmma_f32_16x16x4_f32(false, ayy, false, wfrag[t], (short)0, acc3, false, false);
            acc4 = __builtin_amdgcn_wmma_f32_16x16x4_f32(false, axy, false, wfrag[t], (short)0, acc4, false, false);
        }
        #pragma unroll
        for (int v = 0; v < 8; ++v) {
            int r = mc * TILE + v + 8 * half;   // D layout: M = v (+8 for hi half)
            if (r < PATCH) {
                ht[w][0][r][l16] = acc0[v];
                ht[w][1][r][l16] = acc1[v];
                ht[w][2][r][l16] = acc2[v];
                ht[w][3][r][l16] = acc3[v];
                ht[w][4][r][l16] = acc4[v];
            }
        }
    }
    __syncthreads();

    // ---- vertical pass: Out_p(16x16) = Wv(16x28) * ht_p(28x16) ----
    v8f fr[5];
    #pragma unroll
    for (int p = 0; p < 5; ++p) {
        v8f acc = {};
        #pragma unroll
        for (int t = 0; t < 7; ++t) {
            const int k0 = 4 * t + 2 * half;
            v2f b = {ht[w][p][k0][l16], ht[w][p][k0 + 1][l16]};
            acc = __builtin_amdgcn_wmma_f32_16x16x4_f32(false, wfrag[t], false, b, (short)0, acc, false, false);
        }
        fr[p] = acc;
    }

    // ---- SSIM / CS accumulation over this tile ----
    const float C1 = 4.0e-4f;   // (0.01*2)^2
    const float C2 = 3.6e-3f;   // (0.03*2)^2
    float ssum = 0.f, csum = 0.f;
    #pragma unroll
    for (int v = 0; v < 8; ++v) {
        const int rr = v + 8 * half;
        const int gr = baseR + rr;
        const int gc = baseC + l16;
        const bool valid = active && (gr < Hc) && (gc < Wc);
        float mu1 = fr[0][v], mu2 = fr[1][v];
        float exx = fr[2][v], eyy = fr[3][v], exy = fr[4][v];
        float m11 = mu1 * mu1, m22 = mu2 * mu2, m12 = mu1 * mu2;
        float s1 = exx - m11, s2 = eyy - m22, s12 = exy - m12;
        float v1 = 2.f * s12 + C2;
        float v2 = s1 + s2 + C2;
        float cs = v1 / v2;
        float ss = ((2.f * m12 + C1) * v1) / ((m11 + m22 + C1) * v2);
        if (valid) { ssum += ss; csum += cs; }
    }
    // wave32 reduction (deterministic)
    #pragma unroll
    for (int off = 16; off > 0; off >>= 1) {
        ssum += __shfl_xor(ssum, off, 32);
        csum += __shfl_xor(csum, off, 32);
    }
    if (active && lane == 0) {
        partials[2 * tileFlat]     = ssum;
        partials[2 * tileFlat + 1] = csum;
    }
}

// Deterministic fixed-order reduction of per-tile partials -> acc[0..1]
__global__ __launch_bounds__(256)
void reduce_partials_kernel(const float* __restrict__ part, int n,
                            float* __restrict__ out2) {
    __shared__ float sA[256], sB[256];
    const int tid = threadIdx.x;
    float a = 0.f, b = 0.f;
    for (int i = tid; i < n; i += 256) {
        a += part[2 * i];
        b += part[2 * i + 1];
    }
    sA[tid] = a; sB[tid] = b;
    __syncthreads();
    for (int s = 128; s > 0; s >>= 1) {
        if (tid < s) { sA[tid] += sA[tid + s]; sB[tid] += sB[tid + s]; }
        __syncthreads();
    }
    if (tid == 0) { out2[0] = sA[0]; out2[1] = sB[0]; }
}

// 2x2 average pool for both images of the pyramid level
__global__ __launch_bounds__(256)
void avgpool2_kernel(const float* __restrict__ x, const float* __restrict__ y,
                     float* __restrict__ ox, float* __restrict__ oy,
                     int H, int W) {
    const int Ho = H >> 1, Wo = W >> 1;
    const size_t n = (size_t)BC_TOTAL * Ho * Wo;
    size_t i = (size_t)blockIdx.x * blockDim.x + threadIdx.x;
    if (i >= n) return;
    int wo = (int)(i % Wo);
    size_t t = i / Wo;
    int ho = (int)(t % Ho);
    int bc = (int)(t / Ho);
    size_t base = ((size_t)bc * H + 2 * ho) * W + 2 * wo;
    ox[i] = 0.25f * (x[base] + x[base + 1] + x[base + W] + x[base + W + 1]);
    oy[i] = 0.25f * (y[base] + y[base + 1] + y[base + W] + y[base + W + 1]);
}

__global__ void msssim_finalize_kernel(const float* __restrict__ acc,
                                       float* __restrict__ out) {
    const float wgt[5] = {0.0448f, 0.2856f, 0.3001f, 0.2363f, 0.1333f};
    const int   Hs[5]  = {512, 256, 128, 64, 32};
    float prod = 1.f;
    float csLast = 0.f;
    #pragma unroll
    for (int L = 0; L < 5; ++L) {
        int hc = Hs[L] - 10;
        float cnt = (float)BC_TOTAL * (float)hc * (float)hc;
        float sm = acc[2 * L] / cnt;
        prod *= powf(sm, wgt[L]);
        if (L == 4) csLast = acc[2 * L + 1] / cnt;
    }
    out[0] = 1.f - powf(csLast, wgt[4]) * prod;
}

extern "C" void kernel_launch(void* const* d_in, const int* in_sizes, int n_in,
                              void* d_out, int out_size, void* d_ws, size_t ws_size,
                              hipStream_t stream) {
    (void)in_sizes; (void)n_in; (void)out_size; (void)ws_size;

    const float* X0 = (const float*)d_in[0];
    const float* Y0 = (const float*)d_in[1];
    float* out = (float*)d_out;

    // workspace layout (floats)
    float* accums   = (float*)d_ws;        // 16 floats (10 used)
    float* partials = accums + 16;         // 2 * 49152 floats (level-0 worst case)
    float* pool     = partials + 2 * 49152;

    const size_t n1 = (size_t)BC_TOTAL * 256 * 256;
    const size_t n2 = (size_t)BC_TOTAL * 128 * 128;
    const size_t n3 = (size_t)BC_TOTAL * 64 * 64;
    const size_t n4 = (size_t)BC_TOTAL * 32 * 32;
    float* l1x = pool;            float* l1y = l1x + n1;
    float* l2x = l1y + n1;        float* l2y = l2x + n2;
    float* l3x = l2y + n2;        float* l3y = l3x + n3;
    float* l4x = l3y + n3;        float* l4y = l4x + n4;

    const int Hs[5] = {512, 256, 128, 64, 32};
    const float* xp[5] = {X0, l1x, l2x, l3x, l4x};
    const float* yp[5] = {Y0, l1y, l2y, l3y, l4y};
    float* nxp[5] = {nullptr, l1x, l2x, l3x, l4x};
    float* nyp[5] = {nullptr, l1y, l2y, l3y, l4y};

    for (int L = 0; L < 5; ++L) {
        const int H = Hs[L];
        const int Hc = H - 10;
        const int T = (Hc + TILE - 1) / TILE;
        const int total = BC_TOTAL * T * T;
        const int blocks = (total + WAVES_PER_BLOCK - 1) / WAVES_PER_BLOCK;

        msssim_level_kernel<<<blocks, 128, 0, stream>>>(xp[L], yp[L], partials, H, H);
        reduce_partials_kernel<<<1, 256, 0, stream>>>(partials, total, accums + 2 * L);

        if (L < 4) {
            const size_t np = (size_t)BC_TOTAL * (H / 2) * (H / 2);
            const int pblocks = (int)((np + 255) / 256);
            avgpool2_kernel<<<pblocks, 256, 0, stream>>>(xp[L], yp[L],
                                                         nxp[L + 1], nyp[L + 1], H, H);
        }
    }
    msssim_finalize_kernel<<<1, 1, 0, stream>>>(accums, out);
}